// Sinkhorn_seq_w_29386166239350
// MI455X (gfx1250) — compile-verified
//
#include <hip/hip_runtime.h>
#include <math.h>

// ---------------------------------------------------------------------------
// Sinkhorn (batch element 3 only, per the reference's losses[-1] semantics)
//   B=4, N=512, D=256, EPS=0.1, ITERS=20, LOSS_SCALE=1e-3, STD_EPS=1e-4
// ---------------------------------------------------------------------------

#define N_PTS   512
#define D_DIM   256
#define N_ITERS 20

typedef float v2f __attribute__((ext_vector_type(2)));
typedef float v8f __attribute__((ext_vector_type(8)));

typedef unsigned int u32x4 __attribute__((ext_vector_type(4)));
typedef int          i32x4 __attribute__((ext_vector_type(4)));
typedef int          i32x8 __attribute__((ext_vector_type(8)));

#if defined(__has_builtin)
#if __has_builtin(__builtin_amdgcn_tensor_load_to_lds) && \
    __has_builtin(__builtin_amdgcn_s_wait_tensorcnt)
#define HAVE_TDM 1
#endif
#endif
#ifndef HAVE_TDM
#define HAVE_TDM 0
#endif

// ---------------------------------------------------------------------------
// Kernel 1: z-score normalize rows of batch-3 of y_s / y_t.
// 1024 blocks x 256 threads: block b<512 -> p_s row b, else p_t row b-512.
// Unbiased std (ddof=1), denominator (std + 1e-4), matching jnp.std(ddof=1).
// ---------------------------------------------------------------------------
__global__ void normalize_k(const float* __restrict__ ys,
                            const float* __restrict__ yt,
                            float* __restrict__ ps,
                            float* __restrict__ pt) {
  __shared__ float red[256];
  const int b   = blockIdx.x;
  const int tid = threadIdx.x;

  const float* src;
  float* dst;
  int row;
  if (b < N_PTS) { row = b;          src = ys; dst = ps; }
  else           { row = b - N_PTS;  src = yt; dst = pt; }

  // batch index 3 only
  const size_t off = (size_t)3 * N_PTS * D_DIM + (size_t)row * D_DIM;
  const float v = src[off + tid];

  red[tid] = v;
  __syncthreads();
  for (int s = 128; s > 0; s >>= 1) {
    if (tid < s) red[tid] += red[tid + s];
    __syncthreads();
  }
  const float mean = red[0] * (1.0f / 256.0f);
  __syncthreads();                       // everyone read red[0] before reuse

  const float d = v - mean;
  red[tid] = d * d;
  __syncthreads();
  for (int s = 128; s > 0; s >>= 1) {
    if (tid < s) red[tid] += red[tid + s];
    __syncthreads();
  }
  const float var  = red[0] * (1.0f / 255.0f);   // ddof = 1
  const float stdv = sqrtf(var);
  dst[(size_t)row * D_DIM + tid] = d / (stdv + 1e-4f);
}

// ---------------------------------------------------------------------------
// TDM helper: DMA one 16-row x 256-float tile (row-major, row stride 256
// floats in memory) into LDS with 4 DWORDs of padding after every row, so
// the LDS image has a 260-float row stride (bank-conflict-free reads).
// D# bit layout per CDNA5 ISA ch.8 (groups 0/1; groups 2/3 zero for 2D).
// Issued once per wave (TDM ignores EXEC); completion via TENSORcnt.
// ---------------------------------------------------------------------------
#if HAVE_TDM
__device__ __forceinline__ void tdm_load_tile_16x256(const float* gsrc,
                                                     unsigned lds_byte_off) {
  union { unsigned u[4]; u32x4 v; } g0;
  union { int      i[8]; i32x8 v; } g1;
  const unsigned long long ga = (unsigned long long)(uintptr_t)gsrc;

  // -------- group 0 (128b): count / lds_addr / global_addr / type --------
  g0.u[0] = 1u;                                   // count=1, user descriptor
  g0.u[1] = lds_byte_off;                         // lds_addr (bytes)
  g0.u[2] = (unsigned)(ga & 0xFFFFFFFFu);         // global_addr[31:0]
  g0.u[3] = (unsigned)((ga >> 32) & 0x1FFFFFFu)   // global_addr[56:32]
          | (2u << 30);                           // type = 2 ("image")

  // -------- group 1 (256b): sizes / strides / padding --------
  g1.i[0] = (2 << 16)                             // data_size = 4 bytes
          | (1 << 20)                             // pad_enable
          | (7 << 22)                             // pad_interval: 256 DWORDs
          | (3 << 25);                            // pad_amount: 4 DWORDs
  g1.i[1] = (int)(256u << 16);                    // [63:48]  tensor_dim0 lo16
  g1.i[2] = (int)(512u << 16);                    // [95:80]  tensor_dim1 lo16
  g1.i[3] = (int)(256u << 16);                    // [127:112] tile_dim0 = 256
  g1.i[4] = 16;                                   // [143:128] tile_dim1 = 16
  g1.i[5] = 256;                                  // tensor_dim0_stride lo32
  g1.i[6] = 0;
  g1.i[7] = 0;

  const i32x4 g2 = {0, 0, 0, 0};                  // unused (2D tensor)
  const i32x4 g3 = {0, 0, 0, 0};

#if __clang_major__ >= 23
  const i32x8 g4 = {0, 0, 0, 0, 0, 0, 0, 0};
  __builtin_amdgcn_tensor_load_to_lds(g0.v, g1.v, g2, g3, g4, 0);
#else
  __builtin_amdgcn_tensor_load_to_lds(g0.v, g1.v, g2, g3, 0);
#endif
}
#endif

// ---------------------------------------------------------------------------
// Kernel 2: W[i,j] = sum_d |ps[i,d] - pt[j,d]| ; P[i,j] = exp(-W/0.1).
// Grid (32,32), 256 threads: each block -> one 16x16 tile of W.
// LDS tiles padded to stride 260 (= 64k+4) so per-lane reads across tj hit
// distinct banks (broadcast pairs only). Tiles staged by the Tensor Data
// Mover when available (wave 0 issues 2 DMAs, waits TENSORcnt, barrier).
// ---------------------------------------------------------------------------
#define DSTR 260
__global__ void cdist_k(const float* __restrict__ ps,
                        const float* __restrict__ pt,
                        float* __restrict__ W,
                        float* __restrict__ P) {
  __shared__ float sps[16 * DSTR];
  __shared__ float spt[16 * DSTR];
  const int bi  = blockIdx.x;
  const int bj  = blockIdx.y;
  const int tid = threadIdx.x;

  // confirmed gfx1250 path: emits global_prefetch_b8
  __builtin_prefetch(&ps[(size_t)(bi * 16) * D_DIM + tid], 0, 1);
  __builtin_prefetch(&pt[(size_t)(bj * 16) * D_DIM + tid], 0, 1);

#if HAVE_TDM
  if (tid < 32) {   // wave 0 only; TDM is per-wave (EXEC ignored)
    tdm_load_tile_16x256(&ps[(size_t)(bi * 16) * D_DIM],
                         (unsigned)(unsigned long long)(uintptr_t)sps);
    tdm_load_tile_16x256(&pt[(size_t)(bj * 16) * D_DIM],
                         (unsigned)(unsigned long long)(uintptr_t)spt);
    __builtin_amdgcn_s_wait_tensorcnt(0);
  }
  __syncthreads();
#else
  for (int idx = tid; idx < 16 * D_DIM; idx += 256) {
    const int r = idx >> 8;       // row within tile
    const int d = idx & 255;      // coalesced over d
    sps[r * DSTR + d] = ps[(size_t)(bi * 16 + r) * D_DIM + d];
    spt[r * DSTR + d] = pt[(size_t)(bj * 16 + r) * D_DIM + d];
  }
  __syncthreads();
#endif

  const int ti = tid >> 4;        // lanes vary tj fastest -> coalesced stores
  const int tj = tid & 15;
  const float* a = &sps[ti * DSTR];
  const float* b = &spt[tj * DSTR];
  float acc = 0.0f;
#pragma unroll 8
  for (int d = 0; d < D_DIM; ++d) acc += fabsf(a[d] - b[d]);

  const int i = bi * 16 + ti;
  const int j = bj * 16 + tj;
  W[(size_t)i * N_PTS + j] = acc;
  P[(size_t)i * N_PTS + j] = expf(-acc * 10.0f);   // -W / EPSILON
}

// ---------------------------------------------------------------------------
// Kernel 3: WMMA row/col sums.  sums = P · 1   (TRANS=false -> row sums,
// TRANS=true -> sums over rows == column sums, i.e. rowsum of P^T).
// One wave (32 threads) per 16 output sums; 128 x V_WMMA_F32_16X16X4_F32
// accumulating into an f32 16x16 D whose every column equals the row sums.
// A layout (f32 16x4): lanes 0-15 hold K=k0,k0+1; lanes 16-31 hold K=k0+2,+3.
// C/D layout: VGPR r, lanes 0-15 -> M=r; lanes 16-31 -> M=8+r.
// EXEC is all ones (32 active lanes, no divergence before the WMMA loop).
// ---------------------------------------------------------------------------
template <bool TRANS>
__global__ void sum_wmma_k(const float* __restrict__ P,
                           float* __restrict__ sums) {
  const int lane  = threadIdx.x;        // 0..31
  const int base  = blockIdx.x * 16;    // 16 rows (or cols) per wave
  const int m     = lane & 15;
  const int khalf = (lane >> 4) * 2;    // 0 or 2

  v8f acc = {};
  const v2f ones = {1.0f, 1.0f};        // B = all-ones 4x16, layout-agnostic

  for (int k0 = 0; k0 < N_PTS; k0 += 4) {
    const int k = k0 + khalf;
    v2f a;
    if (TRANS) {
      a.x = P[(size_t)(k + 0) * N_PTS + base + m];
      a.y = P[(size_t)(k + 1) * N_PTS + base + m];
    } else {
      a.x = P[(size_t)(base + m) * N_PTS + k + 0];
      a.y = P[(size_t)(base + m) * N_PTS + k + 1];
    }
    // (neg_a, A, neg_b, B, c_mod, C, reuse_a, reuse_b)
    acc = __builtin_amdgcn_wmma_f32_16x16x4_f32(
        false, a, false, ones, (short)0, acc, false, false);
  }

  // Column 0 of D: lane 0 holds M=0..7 in acc[0..7]; lane 16 holds M=8..15.
  if (lane == 0) {
#pragma unroll
    for (int r = 0; r < 8; ++r) sums[base + r] = acc[r];
  } else if (lane == 16) {
#pragma unroll
    for (int r = 0; r < 8; ++r) sums[base + 8 + r] = acc[r];
  }
}

// ---------------------------------------------------------------------------
// Kernel 4: scale P by 1/sums (COL=false: divide row i by sums[i];
//           COL=true: divide col j by sums[j]).  Fully coalesced.
// ---------------------------------------------------------------------------
template <bool COL>
__global__ void scale_k(float* __restrict__ P, const float* __restrict__ sums) {
  const int idx = blockIdx.x * 256 + threadIdx.x;   // 0 .. 262143
  const int i = idx >> 9;
  const int j = idx & (N_PTS - 1);
  P[idx] /= (COL ? sums[j] : sums[i]);
}

// ---------------------------------------------------------------------------
// Kernel 5: loss = 1e-3 * sum(P * W), two-stage reduction.
// ---------------------------------------------------------------------------
__global__ void loss_partial_k(const float* __restrict__ P,
                               const float* __restrict__ W,
                               float* __restrict__ partial) {
  __shared__ float red[256];
  const int tid  = threadIdx.x;
  const int base = blockIdx.x * 512;
  float acc = P[base + tid] * W[base + tid]
            + P[base + 256 + tid] * W[base + 256 + tid];
  red[tid] = acc;
  __syncthreads();
  for (int s = 128; s > 0; s >>= 1) {
    if (tid < s) red[tid] += red[tid + s];
    __syncthreads();
  }
  if (tid == 0) partial[blockIdx.x] = red[0];
}

__global__ void loss_final_k(const float* __restrict__ partial,
                             float* __restrict__ out) {
  __shared__ float red[256];
  const int tid = threadIdx.x;
  red[tid] = partial[tid] + partial[tid + 256];
  __syncthreads();
  for (int s = 128; s > 0; s >>= 1) {
    if (tid < s) red[tid] += red[tid + s];
    __syncthreads();
  }
  if (tid == 0) out[0] = 1.0e-3f * red[0];
}

// ---------------------------------------------------------------------------
// Launch: normalize -> cdist -> 20 x (rowsum, rowscale, colsum, colscale)
//         -> reduce.  All on `stream`, graph-capture safe.
// ---------------------------------------------------------------------------
extern "C" void kernel_launch(void* const* d_in, const int* in_sizes, int n_in,
                              void* d_out, int out_size, void* d_ws, size_t ws_size,
                              hipStream_t stream) {
  (void)in_sizes; (void)n_in; (void)out_size; (void)ws_size;

  const float* ys = (const float*)d_in[0];
  const float* yt = (const float*)d_in[1];
  float* ws = (float*)d_ws;

  float* ps      = ws;                      // 512*256
  float* pt      = ws + 131072;             // 512*256
  float* W       = ws + 262144;             // 512*512
  float* P       = ws + 524288;             // 512*512
  float* sums    = ws + 786432;             // 512
  float* partial = ws + 786944;             // 512
  float* out     = (float*)d_out;

  normalize_k<<<1024, 256, 0, stream>>>(ys, yt, ps, pt);

  dim3 grid_cd(32, 32);
  cdist_k<<<grid_cd, 256, 0, stream>>>(ps, pt, W, P);

  for (int it = 0; it < N_ITERS; ++it) {
    sum_wmma_k<false><<<32, 32, 0, stream>>>(P, sums);   // row sums
    scale_k<false><<<1024, 256, 0, stream>>>(P, sums);   // P /= rowsum
    sum_wmma_k<true><<<32, 32, 0, stream>>>(P, sums);    // col sums
    scale_k<true><<<1024, 256, 0, stream>>>(P, sums);    // P /= colsum
  }

  loss_partial_k<<<512, 256, 0, stream>>>(P, W, partial);
  loss_final_k<<<1, 256, 0, stream>>>(partial, out);
}